// Post_Attn_26482768347257
// MI455X (gfx1250) — compile-verified
//
#include <hip/hip_runtime.h>
#include <math.h>

// Problem constants (match reference)
#define B_   32
#define N_   4096
#define H_   256
#define NNZ_ 65536
#define CH_  16                 // output-stage chunks per batch
#define ROWS_PER_CH (N_ / CH_)  // 256 rows per chunk

typedef float v2f __attribute__((ext_vector_type(2)));
typedef float v8f __attribute__((ext_vector_type(8)));

// ---------------------------------------------------------------------------
// 0a) clear the valid bitmask (B*N bits = 4096 u32 words)
// ---------------------------------------------------------------------------
__global__ void zero_mask_kernel(unsigned* __restrict__ mask) {
  int i = blockIdx.x * blockDim.x + threadIdx.x;
  if (i < B_ * N_ / 32) mask[i] = 0u;
}

// ---------------------------------------------------------------------------
// 0b) scatter: mark valid (batch,row) positions. atomicOr is idempotent ->
//     bit-exact deterministic final mask regardless of ordering.
// ---------------------------------------------------------------------------
__global__ void scatter_mask_kernel(const int* __restrict__ bi,
                                    const int* __restrict__ ri,
                                    unsigned* __restrict__ mask) {
  int i = blockIdx.x * blockDim.x + threadIdx.x;
  if (i < NNZ_) {
    int bit = bi[i] * N_ + ri[i];
    atomicOr(&mask[bit >> 5], 1u << (bit & 31));
  }
}

// ---------------------------------------------------------------------------
// 0c) c_b = x[b,0,:] . W[H:2H] + bias   (one block of H threads per batch)
// ---------------------------------------------------------------------------
__global__ void cb_kernel(const float* __restrict__ x,
                          const float* __restrict__ W,
                          const float* __restrict__ bias,
                          float* __restrict__ cb) {
  __shared__ float red[H_];
  const int b = blockIdx.x, t = threadIdx.x;
  red[t] = x[(size_t)b * N_ * H_ + t] * W[H_ + t];
  __syncthreads();
  for (int s = H_ / 2; s > 0; s >>= 1) {
    if (t < s) red[t] += red[t + s];
    __syncthreads();
  }
  if (t == 0) cb[b] = red[0] + bias[0];
}

// ---------------------------------------------------------------------------
// 1) scores[b,n] = x[b,n,:] . W[0:H]  via V_WMMA_F32_16X16X4_F32.
//    One wave per 16-row tile; K loop over H in steps of 4.
//    A layout (16x4 f32): lane<16 holds K=0,1 ; lane>=16 holds K=2,3  ->
//    a single aligned v2f load at column k + 2*(lane>>4).
//    B = W_lo broadcast into all 16 columns: VGPR0 = rows K0/K2, VGPR1 = K1/K3.
//    After the loop, D column 0 (lanes 0 and 16) holds the 16 row scores.
// ---------------------------------------------------------------------------
__global__ void scores_wmma_kernel(const float* __restrict__ x,
                                   const float* __restrict__ W,
                                   float* __restrict__ scores) {
  const int lane = threadIdx.x & 31;
  const int wave = threadIdx.x >> 5;
  const int tile = blockIdx.x * (blockDim.x >> 5) + wave;  // 0 .. B*N/16-1
  const int bb   = tile >> 8;                               // tile / (N/16)
  const int n0   = (tile & 255) << 4;                       // row tile base
  const int mrow = lane & 15;
  const int hi   = lane >> 4;                               // 0 or 1

  const float* xrow = x + ((size_t)bb * N_ + (n0 + mrow)) * H_ + 2 * hi;
  const float* wrow = W + 2 * hi;

  v8f acc = {};
#pragma unroll 4
  for (int k = 0; k < H_; k += 4) {
    v2f a  = *(const v2f*)(xrow + k);   // A: x[row, k+2*hi .. k+2*hi+1]
    v2f bv = *(const v2f*)(wrow + k);   // B: W  [k+2*hi .. k+2*hi+1] (all cols)
    acc = __builtin_amdgcn_wmma_f32_16x16x4_f32(
        /*neg_a=*/false, a, /*neg_b=*/false, bv,
        /*c_mod=*/(short)0, acc, /*reuse_a=*/false, /*reuse_b=*/false);
  }

  if (mrow == 0) {  // lanes 0 and 16 carry column 0 of D: M = 8*hi + r
    float* s = scores + (size_t)bb * N_ + n0 + 8 * hi;
#pragma unroll
    for (int r = 0; r < 8; ++r) s[r] = acc[r];
  }
}

// ---------------------------------------------------------------------------
// 2) masked softmax over N per batch. attn = 0 at invalid positions
//    (matches reference: exp(-inf) = 0). One block of 256 threads per batch,
//    16 strided rows per thread; LDS tree reductions for max and sum.
// ---------------------------------------------------------------------------
__global__ void softmax_kernel(const float* __restrict__ scores,
                               const unsigned* __restrict__ mask,
                               const float* __restrict__ cb,
                               float* __restrict__ attn) {
  const int b = blockIdx.x, t = threadIdx.x;
  const float c = cb[b];
  const float NINF = -__builtin_inff();

  float sc[16];
  float mx = NINF;
#pragma unroll
  for (int i = 0; i < 16; ++i) {
    int n = t + i * 256;
    int bit = b * N_ + n;
    bool valid = (mask[bit >> 5] >> (bit & 31)) & 1u;
    float s = valid ? scores[(size_t)b * N_ + n] + c : NINF;
    sc[i] = s;
    mx = fmaxf(mx, s);
  }

  __shared__ float red[256];
  red[t] = mx;
  __syncthreads();
  for (int s = 128; s > 0; s >>= 1) {
    if (t < s) red[t] = fmaxf(red[t], red[t + s]);
    __syncthreads();
  }
  mx = red[0];
  __syncthreads();

  float sum = 0.f;
#pragma unroll
  for (int i = 0; i < 16; ++i) {
    float e = (sc[i] == NINF) ? 0.f : expf(sc[i] - mx);
    sc[i] = e;
    sum += e;
  }
  red[t] = sum;
  __syncthreads();
  for (int s = 128; s > 0; s >>= 1) {
    if (t < s) red[t] += red[t + s];
    __syncthreads();
  }
  const float inv = 1.f / red[0];

#pragma unroll
  for (int i = 0; i < 16; ++i)
    attn[(size_t)b * N_ + t + i * 256] = sc[i] * inv;
}

// ---------------------------------------------------------------------------
// 3a) partial out: block (chunk, batch), thread t owns column h=t.
//     Skip rows with attn==0 (uniform branch per block) -> ~2x less HBM.
// ---------------------------------------------------------------------------
__global__ void out_partial_kernel(const float* __restrict__ x,
                                   const float* __restrict__ attn,
                                   float* __restrict__ partial) {
  const int ch = blockIdx.x, b = blockIdx.y, t = threadIdx.x;
  const int nbeg = ch * ROWS_PER_CH;
  const float* xb = x + (size_t)b * N_ * H_;
  const float* ab = attn + (size_t)b * N_;

  float acc = 0.f;
  for (int n = nbeg; n < nbeg + ROWS_PER_CH; ++n) {
    float a = ab[n];
    if (a != 0.f) acc = fmaf(a, xb[(size_t)n * H_ + t], acc);
  }
  partial[((size_t)b * CH_ + ch) * H_ + t] = acc;
}

// ---------------------------------------------------------------------------
// 3b) fixed-order chunk reduction -> deterministic out[b,h]
// ---------------------------------------------------------------------------
__global__ void out_reduce_kernel(const float* __restrict__ partial,
                                  float* __restrict__ out) {
  const int b = blockIdx.x, t = threadIdx.x;
  float s = 0.f;
#pragma unroll
  for (int ch = 0; ch < CH_; ++ch)
    s += partial[((size_t)b * CH_ + ch) * H_ + t];
  out[(size_t)b * H_ + t] = s;
}

// ---------------------------------------------------------------------------
extern "C" void kernel_launch(void* const* d_in, const int* in_sizes, int n_in,
                              void* d_out, int out_size, void* d_ws, size_t ws_size,
                              hipStream_t stream) {
  const float* x  = (const float*)d_in[0];
  const int*   bi = (const int*)d_in[1];
  const int*   ri = (const int*)d_in[2];
  const float* W  = (const float*)d_in[3];
  const float* bs = (const float*)d_in[4];  // scalar bias

  // Workspace layout (floats / u32 words), ~1.06 MB total:
  unsigned* mask   = (unsigned*)d_ws;                 // 4096 words (16 KB)
  float*    cb     = (float*)d_ws + 4096;             // 32
  float*    scores = cb + 32;                         // B*N = 131072
  float*    partial= scores + (size_t)B_ * N_;        // B*CH*H = 131072

  float* out  = (float*)d_out;                        // [B,H]   = 8192
  float* attn = out + (size_t)B_ * H_;                // [B,N,1] = 131072

  zero_mask_kernel<<<(B_ * N_ / 32 + 255) / 256, 256, 0, stream>>>(mask);
  scatter_mask_kernel<<<NNZ_ / 256, 256, 0, stream>>>(bi, ri, mask);
  cb_kernel<<<B_, H_, 0, stream>>>(x, W, bs, cb);
  scores_wmma_kernel<<<(B_ * (N_ / 16)) / 8, 256, 0, stream>>>(x, W, scores);
  softmax_kernel<<<B_, 256, 0, stream>>>(scores, mask, cb, attn);
  out_partial_kernel<<<dim3(CH_, B_), H_, 0, stream>>>(x, attn, partial);
  out_reduce_kernel<<<B_, H_, 0, stream>>>(partial, out);
}